// WKA_44117904065351
// MI455X (gfx1250) — compile-verified
//
#include <hip/hip_runtime.h>
#include <math.h>

// ---------------------------------------------------------------------------
// WKA (omega-K SAR) pipeline for MI455X / gfx1250, wave32.
//   K1  row FFT (fwd)  sig -> o0                  [m][k]
//   K2  transpose      o0 -> o1                   [k][m]
//   K3  row FFT (fwd)  o1 -> o2  (= s2df^T)       [k][l]
//   K4  transpose+match o2 -> o0 (s2df), o1 (s2df*H2DF)   [l][k]
//   K5  stolt          o1 -> o2  (s2df_stolt, flipped)    [l][k]
//   K6  row FFT (inv)  o2 -> o3                   [l][k']
//   K7  transpose      o3 -> ws                   [k'][l]
//   K8  row FFT (inv)  ws -> ws (in place)        [k'][l'] = image^T
//   K9  transpose      ws -> o3  (image)          [l'][k']
// fftshift pairs folded into (-1)^n input / (-1)^k output signs (exact, even N).
// 2048-pt FFT = radix-16 stage on the WMMA f32 matrix pipe + 7-stage Stockham
// radix-2 in LDS.  Row staging uses GLOBAL_LOAD_ASYNC_TO_LDS_B128 (ASYNCcnt).
// ---------------------------------------------------------------------------

#define NSZ   2048
#define NN    (NSZ * NSZ)
#define SSTR  130          // padded segment stride (bank-conflict-free readout)

typedef __attribute__((ext_vector_type(2))) float v2f;
typedef __attribute__((ext_vector_type(8))) float v8f;

#if defined(__has_builtin)
#  if __has_builtin(__builtin_amdgcn_global_load_async_to_lds_b128) && \
      __has_builtin(__builtin_amdgcn_s_wait_asynccnt)
#    define HAVE_ASYNC_LDS 1
#  endif
#endif
#ifndef HAVE_ASYNC_LDS
#  define HAVE_ASYNC_LDS 0
#endif

// Builtin expects pointers to 16B int vectors (per hipcc diagnostic):
//   param0: '__attribute__((__vector_size__(4*sizeof(int)))) int __device__ *'
typedef int v4i_vs __attribute__((__vector_size__(16)));
typedef __attribute__((address_space(1))) v4i_vs gv4i_t;  // global
typedef __attribute__((address_space(3))) v4i_vs lv4i_t;  // LDS

struct alignas(16) FftLds {
  float2 x[16 * SSTR];
  float2 y[16 * SSTR];
};

static __device__ __forceinline__ v8f wmma4(v2f a, v2f b, v8f c) {
  // D(16x16,f32) += A(16x4,f32) * B(4x16,f32); no A/B neg for f32 (ISA 7.12)
  return __builtin_amdgcn_wmma_f32_16x16x4_f32(false, a, false, b,
                                               (short)0, c, false, false);
}

// 2048-pt complex FFT of raw data in L.x (natural order); input fftshift sign
// (-1)^n is applied internally (== (-1)^{n2}).  dir=-1 fwd / +1 inv.
// Result left in L.x at padded layout: X[k2*16+k1] = L.x[k1*SSTR + k2].
static __device__ void fft2048_core(FftLds& L, int tid, float dir) {
  const int lane = tid & 31;
  const int wv   = tid >> 5;           // wave = tile index t (0..7)
  const int colL = lane & 15;
  const int kb   = (lane >> 4) << 1;   // 0 (lanes 0-15) or 2 (lanes 16-31)
  const int half = lane >> 4;
  const int n2   = wv * 16 + colL;     // column of the 16x128 stage-1 matrix
  const float sg2 = (n2 & 1) ? -1.f : 1.f;  // (-1)^n input sign, n == n2 mod 2

  __syncthreads();  // raw input resident in L.x

  // ---- stage 1 (radix-16) on the matrix pipe: Y = W16 * X, complex ----
  v8f cre = {0.f,0.f,0.f,0.f,0.f,0.f,0.f,0.f};
  v8f cim = {0.f,0.f,0.f,0.f,0.f,0.f,0.f,0.f};
  #pragma unroll
  for (int c = 0; c < 4; ++c) {
    const int k0 = 4 * c + kb, k1 = k0 + 1;
    const int m  = colL;  // A row (M = lane&15 in both lane halves)
    float s0, c0s, s1, c1s;
    __sincosf((float)((m * k0) & 15) * 0.39269908169872414f, &s0, &c0s); // pi/8
    __sincosf((float)((m * k1) & 15) * 0.39269908169872414f, &s1, &c1s);
    v2f Ar = { c0s, c1s };                 // Re W16
    v2f Ai = { dir * s0, dir * s1 };       // Im W16 (dir folds conj)
    v2f An = { -dir * s0, -dir * s1 };     // -Im W16
    const float2 b0 = L.x[k0 * 128 + n2];
    const float2 b1 = L.x[k1 * 128 + n2];
    v2f Br = { b0.x, b1.x };
    v2f Bi = { b0.y, b1.y };
    cre = wmma4(Ar, Br, cre);
    cre = wmma4(An, Bi, cre);
    cim = wmma4(Ar, Bi, cim);
    cim = wmma4(Ai, Br, cim);
  }
  // input sign (diagonal over n2, commutes with W16) + twiddle W_N^{n2*k1};
  // scatter to segment-major padded layout in y
  #pragma unroll
  for (int r = 0; r < 8; ++r) {
    const int mm = r + 8 * half;            // k1 (C/D: VGPR r -> rows r, r+8)
    float sw, cw;
    __sincosf(dir * 0.0030679615757712823f * (float)(mm * n2), &sw, &cw); // 2pi/2048
    const float vr = cre[r] * sg2, vi = cim[r] * sg2;
    L.y[mm * SSTR + n2] = make_float2(vr * cw - vi * sw, vr * sw + vi * cw);
  }

  // ---- stage 2: sixteen 128-pt Stockham radix-2 FFTs (7 stages, y<->x) ----
  float2* s_ = L.y;
  float2* d_ = L.x;
  for (int st = 0; st < 7; ++st) {
    __syncthreads();
    const int   nt   = 128 >> st;
    const int   s    = 1 << st;
    const float cang = dir * 6.283185307179586f / (float)nt;
    #pragma unroll
    for (int u = 0; u < 4; ++u) {
      const int g    = tid + u * 256;       // 1024 butterflies
      const int seg  = g >> 6;
      const int j    = g & 63;
      const int p    = j >> st;
      const int q    = j & (s - 1);
      const int base = seg * SSTR;
      const int ia   = base + j, ib = ia + 64;
      const float2 a = s_[ia];
      const float2 b = s_[ib];
      const float er = a.x - b.x, ei = a.y - b.y;
      float sw, cw;
      __sincosf(cang * (float)p, &sw, &cw);
      const int io0 = base + q + (p << (st + 1));
      const int io1 = io0 + s;
      d_[io0] = make_float2(a.x + b.x, a.y + b.y);
      d_[io1] = make_float2(er * cw - ei * sw, er * sw + ei * cw);
    }
    float2* t = s_; s_ = d_; d_ = t;
  }
  __syncthreads();  // 7 stages (odd) -> result in L.x
}

__global__ void __launch_bounds__(256)
fft_rows_planar(const float* __restrict__ re, const float* __restrict__ im,
                float2* __restrict__ out, float dir, float scale) {
  __shared__ FftLds L;
  const int tid = threadIdx.x;
  const int row = blockIdx.x;
  const float* r0 = re + (size_t)row * NSZ;
  const float* i0 = im + (size_t)row * NSZ;
  #pragma unroll
  for (int u = 0; u < 8; ++u) {
    const int n = tid + u * 256;
    L.x[n] = make_float2(r0[n], i0[n]);   // raw; core applies input sign
  }
  fft2048_core(L, tid, dir);
  float2* o = out + (size_t)row * NSZ;
  #pragma unroll
  for (int u = 0; u < 8; ++u) {
    const int k  = tid + u * 256;
    const int k1 = k & 15, k2 = k >> 4;
    const float sg = (k & 1) ? -scale : scale;  // output fftshift sign
    const float2 v = L.x[k1 * SSTR + k2];
    o[k] = make_float2(v.x * sg, v.y * sg);
  }
}

__global__ void __launch_bounds__(256)
fft_rows_c(const float2* in, float2* out, float dir, float scale) {
  __shared__ FftLds L;
  const int tid = threadIdx.x;
  const int row = blockIdx.x;
  const float2* p = in + (size_t)row * NSZ;
#if HAVE_ASYNC_LDS
  // Stage the 16KB row straight into LDS on the ASYNCcnt path (no VGPR trip).
  #pragma unroll
  for (int u = 0; u < 4; ++u) {
    const int e = (tid + u * 256) * 2;        // 2 float2 per b128, 16B aligned
    __builtin_amdgcn_global_load_async_to_lds_b128(
        (gv4i_t*)(p + e), (lv4i_t*)(&L.x[e]), 0, 0);
  }
  __builtin_amdgcn_s_wait_asynccnt(0);
#else
  #pragma unroll
  for (int u = 0; u < 8; ++u) {
    const int n = tid + u * 256;
    L.x[n] = p[n];
  }
#endif
  fft2048_core(L, tid, dir);
  float2* o = out + (size_t)row * NSZ;
  #pragma unroll
  for (int u = 0; u < 8; ++u) {
    const int k  = tid + u * 256;
    const int k1 = k & 15, k2 = k >> 4;
    const float sg = (k & 1) ? -scale : scale;
    const float2 v = L.x[k1 * SSTR + k2];
    o[k] = make_float2(v.x * sg, v.y * sg);
  }
}

__global__ void __launch_bounds__(256)
transpose_k(const float2* __restrict__ in, float2* __restrict__ out) {
  __shared__ float2 tile[32][33];
  const int bx = blockIdx.x * 32, by = blockIdx.y * 32;
  const int tx = threadIdx.x,     ty = threadIdx.y;
  #pragma unroll
  for (int i = 0; i < 32; i += 8)
    tile[ty + i][tx] = in[(size_t)(by + ty + i) * NSZ + bx + tx];
  __syncthreads();
  #pragma unroll
  for (int i = 0; i < 32; i += 8)
    out[(size_t)(bx + ty + i) * NSZ + by + tx] = tile[tx][ty + i];
}

// in = s2df^T ([kr][l]); writes s2df ([l][kr]) and s2df*H2DF, H2DF in f64.
__global__ void __launch_bounds__(256)
transpose_match_k(const float2* __restrict__ in,
                  float2* __restrict__ s2df_o, float2* __restrict__ match_o) {
  __shared__ float2 tile[32][33];
  const int bx = blockIdx.x * 32;   // l blocks
  const int by = blockIdx.y * 32;   // kr blocks
  const int tx = threadIdx.x, ty = threadIdx.y;
  #pragma unroll
  for (int i = 0; i < 32; i += 8)
    tile[ty + i][tx] = in[(size_t)(by + ty + i) * NSZ + bx + tx];
  __syncthreads();
  #pragma unroll
  for (int i = 0; i < 32; i += 8) {
    const int l  = bx + ty + i;
    const int kr = by + tx;
    const float2 v = tile[tx][ty + i];
    s2df_o[(size_t)l * NSZ + kr] = v;
    const double fr   = (double)(kr - 1024) * (32317000.0 / 2048.0);
    const double fa   = (double)(l  - 1024) * (1256.98 / 2048.0) - 6900.0;
    const double fcfr = 5300000000.0 + fr;
    const double arg  = fcfr * fcfr
                      - (299790000.0 * 299790000.0) * fa * fa
                        / (4.0 * 7062.0 * 7062.0);
    const double ph   = (4.0 * M_PI * 850000.0 / 299790000.0) * sqrt(arg)
                      + M_PI * fr * fr / 721350000000.0;
    double sp, cp;
    sincos(ph, &sp, &cp);
    const float hr = (float)cp, hi = (float)sp;
    match_o[(size_t)l * NSZ + kr] =
        make_float2(v.x * hr - v.y * hi, v.x * hi + v.y * hr);
  }
}

// 8-tap Stolt sinc interpolation with circular indexing + both-axis flip.
__global__ void __launch_bounds__(256)
stolt_k(const float2* __restrict__ sm, float2* __restrict__ out) {
  const int n = blockIdx.x * 256 + threadIdx.x;  // range bin
  const int m = blockIdx.y;                      // azimuth bin
  const double fr  = (double)(n - 1024) * (32317000.0 / 2048.0);
  const double fa  = (double)(m - 1024) * (1256.98 / 2048.0) - 6900.0;
  const double a   = fr + 5300000000.0;
  const double b   = fa * (299790000.0 / (2.0 * 7062.0));
  const double Dfr = sqrt(a * a + b * b) - a;
  const double v   = Dfr * (2048.0 / 32317000.0);
  const double nd  = nearbyint(v);               // RNE, matches np.round
  const int    Nd  = (int)nd;
  const float  deci = (float)(v - nd);
  const float2* row = sm + (size_t)m * NSZ;
  float ar = 0.f, ai = 0.f;
  #pragma unroll
  for (int t = 0; t < 8; ++t) {
    const int col = (n + Nd - 5 + t) & (NSZ - 1);  // Nd >= 0 always
    const float2 d = row[col];
    const float x  = deci + (float)(3 - t);
    const float px = x * 3.14159265358979f;
    const float h  = (fabsf(x) < 1e-7f) ? 1.f : (sinf(px) / px);
    ar += h * d.x;      // sum h * conj(d)
    ai -= h * d.y;
  }
  if (n < 7) { ar = 0.f; ai = 0.f; }               // PI_TAPS-1 guard
  out[(size_t)(NSZ - 1 - m) * NSZ + (NSZ - 1 - n)] = make_float2(ar, ai);
}

extern "C" void kernel_launch(void* const* d_in, const int* in_sizes, int n_in,
                              void* d_out, int out_size, void* d_ws, size_t ws_size,
                              hipStream_t stream) {
  (void)in_sizes; (void)n_in; (void)out_size; (void)ws_size;
  const float* sre = (const float*)d_in[0];
  const float* sim = (const float*)d_in[1];
  float2* o0 = (float2*)d_out;        // s2df
  float2* o1 = o0 + NN;               // s2df_matched
  float2* o2 = o1 + NN;               // s2df_stolt
  float2* o3 = o2 + NN;               // image
  float2* W  = (float2*)d_ws;         // one NSZ*NSZ float2 ping buffer

  const dim3 tb(32, 8), tg(NSZ / 32, NSZ / 32);

  fft_rows_planar  <<<NSZ, 256, 0, stream>>>(sre, sim, o0, -1.f, 1.f);
  transpose_k      <<<tg, tb, 0, stream>>>(o0, o1);
  fft_rows_c       <<<NSZ, 256, 0, stream>>>(o1, o2, -1.f, 1.f);
  transpose_match_k<<<tg, tb, 0, stream>>>(o2, o0, o1);
  stolt_k          <<<dim3(NSZ / 256, NSZ), 256, 0, stream>>>(o1, o2);
  fft_rows_c       <<<NSZ, 256, 0, stream>>>(o2, o3, 1.f, 1.f / 2048.f);
  transpose_k      <<<tg, tb, 0, stream>>>(o3, W);
  fft_rows_c       <<<NSZ, 256, 0, stream>>>(W, W, 1.f, 1.f / 2048.f);
  transpose_k      <<<tg, tb, 0, stream>>>(W, o3);
}